// StochasticDownsampling2D_53764400611978
// MI455X (gfx1250) — compile-verified
//
#include <hip/hip_runtime.h>
#include <stdint.h>

// Stochastic 2x downsample gather: out[n,i,j,c] = in[n, rows[i], cols[j], c]
// N=8, H=W=512, C=64 fp32.  Pure data movement: 128 MiB read + 128 MiB write
// -> ~11.5 us at 23.3 TB/s.  Strategy: 16 lanes cover one 256B output pixel
// (b128 per lane), moved through the CDNA5 async global->LDS->global engine
// (ASYNCcnt) so the data never occupies VGPRs.

#define BLOCK 256
#define ITEMS 4   // float4 chunks per thread -> 64 B per thread per launch

typedef float v4f __attribute__((ext_vector_type(4)));

__global__ __launch_bounds__(BLOCK) void sdown_async_kernel(
    const float* __restrict__ in, const int* __restrict__ rows,
    const int* __restrict__ cols, float* __restrict__ out)
{
    // One private 16B LDS slot per (thread, item): 256*4*16 = 16 KiB / block.
    __shared__ alignas(16) unsigned char lds_buf[BLOCK * ITEMS * 16];
    (void)lds_buf;

    const uint32_t tid   = threadIdx.x;
    const uint32_t gbase = blockIdx.x * (BLOCK * ITEMS) + tid;

    uint64_t in_addr[ITEMS];
    uint64_t out_addr[ITEMS];
    uint32_t lds_off[ITEMS];

#pragma unroll
    for (int k = 0; k < ITEMS; ++k) {
        // g indexes float4 chunks of the output [8, 256, 256, 16] (all pow2).
        const uint32_t g  = gbase + (uint32_t)k * BLOCK;
        const uint32_t c4 = g & 15u;          // float4 within channel dim
        const uint32_t j  = (g >> 4)  & 255u; // out col
        const uint32_t i  = (g >> 12) & 255u; // out row
        const uint32_t n  = g >> 20;          // batch
        const uint32_t r  = (uint32_t)rows[i];
        const uint32_t c  = (uint32_t)cols[j];
        // input flat float index: ((n*512 + r)*512 + c)*64 + c4*4
        const uint64_t in_elem = ((uint64_t)n << 24) + ((uint64_t)r << 15) +
                                 ((uint64_t)c << 6)  + ((uint64_t)(c4 << 2));
        in_addr[k]  = (uint64_t)(uintptr_t)in  + (in_elem << 2);
        out_addr[k] = (uint64_t)(uintptr_t)out + ((uint64_t)g << 4);
        lds_off[k]  = ((uint32_t)k * BLOCK + tid) * 16u;
    }

    // Issue all 4 async gathers (ASYNCcnt), wait once, then 4 async stores.
#pragma unroll
    for (int k = 0; k < ITEMS; ++k)
        asm volatile("global_load_async_to_lds_b128 %0, %1, off"
                     :: "v"(lds_off[k]), "v"(in_addr[k]) : "memory");
    asm volatile("s_wait_asynccnt 0x0" ::: "memory");
#pragma unroll
    for (int k = 0; k < ITEMS; ++k)
        asm volatile("global_store_async_from_lds_b128 %0, %1, off"
                     :: "v"(out_addr[k]), "v"(lds_off[k]) : "memory");
    asm volatile("s_wait_asynccnt 0x0" ::: "memory");
}

// Fallback (unreferenced): straight VGPR float4 copy, same addressing.
__global__ __launch_bounds__(BLOCK) void sdown_direct_kernel(
    const float* __restrict__ in, const int* __restrict__ rows,
    const int* __restrict__ cols, float* __restrict__ out)
{
    const uint32_t g  = blockIdx.x * BLOCK + threadIdx.x;
    const uint32_t c4 = g & 15u;
    const uint32_t j  = (g >> 4)  & 255u;
    const uint32_t i  = (g >> 12) & 255u;
    const uint32_t n  = g >> 20;
    const uint32_t r  = (uint32_t)rows[i];
    const uint32_t c  = (uint32_t)cols[j];
    const uint64_t in_elem = ((uint64_t)n << 24) + ((uint64_t)r << 15) +
                             ((uint64_t)c << 6)  + ((uint64_t)(c4 << 2));
    const v4f v = __builtin_nontemporal_load((const v4f*)(in + in_elem));
    __builtin_nontemporal_store(v, (v4f*)(out) + g);
}

extern "C" void kernel_launch(void* const* d_in, const int* in_sizes, int n_in,
                              void* d_out, int out_size, void* d_ws, size_t ws_size,
                              hipStream_t stream) {
    (void)n_in; (void)d_ws; (void)ws_size; (void)in_sizes;
    const float* in  = (const float*)d_in[0];
    const int* rows  = (const int*)d_in[1];
    const int* cols  = (const int*)d_in[2];
    float* out       = (float*)d_out;

    // out_size = 8*256*256*64 = 33,554,432 floats = 8,388,608 float4 chunks.
    const int f4_total = out_size / 4;
    const int blocks   = f4_total / (BLOCK * ITEMS);   // 8192
    sdown_async_kernel<<<blocks, BLOCK, 0, stream>>>(in, rows, cols, out);
}